// MixedOp_35562329211102
// MI455X (gfx1250) — compile-verified
//
#include <hip/hip_runtime.h>
#include <math.h>

#define Bn 256
#define Cn 16
#define DIN 64
#define DOUT 32
#define NOPS 6
#define LDM 33
#define MSZ (32*LDM)        // 1056 floats per padded 32x32 matrix
#define LDX 65
#define SQRT_LEVELS 3
#define NS_ITERS 5
#define EXP_SQUARINGS 6

typedef float v2f __attribute__((ext_vector_type(2)));
typedef float v8f __attribute__((ext_vector_type(8)));

__device__ __forceinline__ v8f wmma4(v2f a, v2f b, v8f c) {
  // V_WMMA_F32_16X16X4_F32 : D = A(16x4) * B(4x16) + C
  return __builtin_amdgcn_wmma_f32_16x16x4_f32(false, a, false, b, (short)0, c,
                                               false, false);
}

// A fragment (16x4 tile at row0,k0), row-major source, 32-bit layout per ISA 7.12.2:
// lanes 0-15: M=lane, K={k0,k0+1}; lanes 16-31: M=lane-16, K={k0+2,k0+3}
__device__ __forceinline__ v2f load_a(const float* m, int ld, int row0, int k0, int lane) {
  const int r = row0 + (lane & 15);
  const int k = k0 + ((lane >> 4) << 1);
  v2f a; a.x = m[r*ld + k]; a.y = m[r*ld + k + 1]; return a;
}
// A fragment of M^T: logical A[mrow][k] = M[k][mrow]
__device__ __forceinline__ v2f load_at(const float* m, int ld, int row0, int k0, int lane) {
  const int r = row0 + (lane & 15);
  const int k = k0 + ((lane >> 4) << 1);
  v2f a; a.x = m[k*ld + r]; a.y = m[(k+1)*ld + r]; return a;
}
// B fragment (4x16 tile at k0,col0): lanes 0-15: N=lane, K={k0,k0+1}; lanes 16-31: K={k0+2,k0+3}
__device__ __forceinline__ v2f load_b(const float* m, int ld, int k0, int col0, int lane) {
  const int c = col0 + (lane & 15);
  const int k = k0 + ((lane >> 4) << 1);
  v2f b; b.x = m[k*ld + c]; b.y = m[(k+1)*ld + c]; return b;
}

template<int K>
__device__ __forceinline__ v8f mm_acc(const float* A, int lda, const float* Bm, int ldb,
                                      int row0, int col0, int lane, bool atrans) {
  v8f c = {};
  #pragma unroll
  for (int k0 = 0; k0 < K; k0 += 4) {
    v2f a = atrans ? load_at(A, lda, row0, k0, lane) : load_a(A, lda, row0, k0, lane);
    v2f b = load_b(Bm, ldb, k0, col0, lane);
    c = wmma4(a, b, c);
  }
  return c;
}

// C/D tile layout: VGPR r -> row = row0 + r + 8*(lane>=16), col = col0 + lane%16
__device__ __forceinline__ void store_plain(float* m, int ld, int row0, int col0,
                                            int lane, v8f c) {
  const int col = col0 + (lane & 15);
  const int rb  = row0 + ((lane >> 4) << 3);
  #pragma unroll
  for (int r = 0; r < 8; ++r) m[(rb + r)*ld + col] = c[r];
}
// m = scale*c + diag*I
__device__ __forceinline__ void store_affine(float* m, int ld, int row0, int col0,
                                             int lane, v8f c, float scale, float diag) {
  const int col = col0 + (lane & 15);
  const int rb  = row0 + ((lane >> 4) << 3);
  #pragma unroll
  for (int r = 0; r < 8; ++r) {
    const int row = rb + r;
    m[row*ld + col] = scale*c[r] + ((row == col) ? diag : 0.f);
  }
}

__global__ __launch_bounds__(256)
void mixedop_spd_kernel(const float* __restrict__ x, const float* __restrict__ W,
                        const float* __restrict__ wts, float* __restrict__ out) {
  __shared__ float sA[NOPS*MSZ];   // Y -> normalized A -> NS ping -> exp ping
  __shared__ float sB[NOPS*MSZ];   // NS pong / exp pong
  __shared__ float sC[NOPS*MSZ];   // (phase1: x staging)  Z ping / series ping
  __shared__ float sD[NOPS*MSZ];   // (phase1: W,T staging) Z pong / series pong
  __shared__ float sT[NOPS*MSZ];   // NS "T" matrices; later M and scaled-M
  __shared__ float tr6[NOPS], itr6[NOPS], ltr6[NOPS], wv[NOPS];

  const int tid  = threadIdx.x;
  const int lane = tid & 31;
  const int wave = tid >> 5;
  const int bc   = blockIdx.x;       // b*Cn + c
  const int c    = bc & (Cn - 1);

  float* xs = sC;              // 64*65 = 4160 floats
  float* Ws = sD;              // 64*33 = 2112
  float* Ts = sD + DIN*LDM;    // 64*33

  // ---- stage x[b,c] (16 KB) into LDS, reused by all 6 congruences ----
  const float* xg = x + (size_t)bc * (DIN*DIN);
  for (int i = tid; i < DIN*DIN; i += 256)
    xs[(i >> 6)*LDX + (i & 63)] = xg[i];
  if (tid < NOPS) { tr6[tid] = 0.f; wv[tid] = wts[tid]; }
  __syncthreads();

  // ---- Phase 1: Y[o] = W_o^T X W_o (all WMMA) ----
  for (int o = 0; o < NOPS; ++o) {
    const float* Wg = W + (size_t)(o*Cn + c) * (DIN*DOUT);
    for (int i = tid; i < DIN*DOUT; i += 256)
      Ws[(i >> 5)*LDM + (i & 31)] = Wg[i];
    __syncthreads();
    { // T = X @ W : 64x32, 8 tiles, one per wave
      const int trow = (wave >> 1) * 16, tcol = (wave & 1) * 16;
      v8f acc = mm_acc<DIN>(xs, LDX, Ws, LDM, trow, tcol, lane, false);
      store_plain(Ts, LDM, trow, tcol, lane, acc);
    }
    __syncthreads();
    if (wave < 4) { // Y = W^T @ T : 32x32, 4 tiles
      const int trow = (wave >> 1) * 16, tcol = (wave & 1) * 16;
      v8f acc = mm_acc<DIN>(Ws, LDM, Ts, LDM, trow, tcol, lane, true);
      store_plain(sA + o*MSZ, LDM, trow, tcol, lane, acc);
    }
    __syncthreads();
  }

  // ---- Phase 2: trace, log(trace), then in-place symmetrize + trace-normalize ----
  for (int j = tid; j < NOPS*32; j += 256)
    atomicAdd(&tr6[j >> 5], sA[(j >> 5)*MSZ + (j & 31)*LDM + (j & 31)]);
  __syncthreads();
  if (tid < NOPS) {
    float t = fmaxf(tr6[tid], 1e-20f);
    itr6[tid] = 1.f / t;
    ltr6[tid] = logf(t);
  }
  __syncthreads();
  for (int j = tid; j < NOPS*1024; j += 256) {
    const int m = j >> 10, e = j & 1023, i = e >> 5, jj = e & 31;
    if (i <= jj) {  // each (i,j)/(j,i) pair owned by exactly one thread
      float* Ym = sA + m*MSZ;
      const float v = 0.5f*(Ym[i*LDM + jj] + Ym[jj*LDM + i]) * itr6[m];
      Ym[i*LDM + jj] = v;
      Ym[jj*LDM + i] = v;
    }
  }
  __syncthreads();

  // ---- Phase 3: A^(1/2^s) via coupled Newton-Schulz, 6 matrices concurrently ----
  float* Ycur = sA; float* Yalt = sB;
  float* Zcur = sC; float* Zalt = sD;
  for (int lvl = 0; lvl < SQRT_LEVELS; ++lvl) {
    for (int j = tid; j < NOPS*1024; j += 256) {   // Z = I
      const int m = j >> 10, e = j & 1023, i = e >> 5, jj = e & 31;
      Zcur[m*MSZ + i*LDM + jj] = (i == jj) ? 1.f : 0.f;
    }
    __syncthreads();
    for (int it = 0; it < NS_ITERS; ++it) {
      // T = 1.5 I - 0.5 Z@Y : 24 tile-jobs over 8 waves
      for (int job = wave; job < NOPS*4; job += 8) {
        const int m = job >> 2, t = job & 3;
        const int trow = (t >> 1)*16, tcol = (t & 1)*16;
        v8f acc = mm_acc<32>(Zcur + m*MSZ, LDM, Ycur + m*MSZ, LDM, trow, tcol, lane, false);
        store_affine(sT + m*MSZ, LDM, trow, tcol, lane, acc, -0.5f, 1.5f);
      }
      __syncthreads();
      // Ynew = Y@T ; Znew = T@Z : 48 tile-jobs over 8 waves
      for (int job = wave; job < NOPS*8; job += 8) {
        const int m = job >> 3, t = job & 7;
        const int trow = ((t >> 1) & 1)*16, tcol = (t & 1)*16;
        if (t < 4) {
          v8f acc = mm_acc<32>(Ycur + m*MSZ, LDM, sT + m*MSZ, LDM, trow, tcol, lane, false);
          store_plain(Yalt + m*MSZ, LDM, trow, tcol, lane, acc);
        } else {
          v8f acc = mm_acc<32>(sT + m*MSZ, LDM, Zcur + m*MSZ, LDM, trow, tcol, lane, false);
          store_plain(Zalt + m*MSZ, LDM, trow, tcol, lane, acc);
        }
      }
      __syncthreads();
      float* tmp = Ycur; Ycur = Yalt; Yalt = tmp;
      tmp = Zcur; Zcur = Zalt; Zalt = tmp;
    }
  }

  // ---- E = R - I, then log(I+E) via degree-6 Horner (5 matmuls) ----
  for (int j = tid; j < NOPS*32; j += 256)
    Ycur[(j >> 5)*MSZ + (j & 31)*LDM + (j & 31)] -= 1.f;
  __syncthreads();
  for (int j = tid; j < NOPS*1024; j += 256) {   // H = c5*I + c6*E
    const int m = j >> 10, e = j & 1023, i = e >> 5, jj = e & 31;
    Zcur[m*MSZ + i*LDM + jj] =
        (-1.f/6.f)*Ycur[m*MSZ + i*LDM + jj] + ((i == jj) ? 0.2f : 0.f);
  }
  __syncthreads();
  const float coef[4] = {-0.25f, 1.f/3.f, -0.5f, 1.f};   // c4,c3,c2,c1
  for (int k = 0; k < 4; ++k) {
    for (int job = wave; job < NOPS*4; job += 8) {       // H = ck*I + E@H
      const int m = job >> 2, t = job & 3;
      const int trow = (t >> 1)*16, tcol = (t & 1)*16;
      v8f acc = mm_acc<32>(Ycur + m*MSZ, LDM, Zcur + m*MSZ, LDM, trow, tcol, lane, false);
      store_affine(Zalt + m*MSZ, LDM, trow, tcol, lane, acc, 1.f, coef[k]);
    }
    __syncthreads();
    float* tmp = Zcur; Zcur = Zalt; Zalt = tmp;
  }
  for (int job = wave; job < NOPS*4; job += 8) {         // L = E@H
    const int m = job >> 2, t = job & 3;
    const int trow = (t >> 1)*16, tcol = (t & 1)*16;
    v8f acc = mm_acc<32>(Ycur + m*MSZ, LDM, Zcur + m*MSZ, LDM, trow, tcol, lane, false);
    store_plain(Zalt + m*MSZ, LDM, trow, tcol, lane, acc);
  }
  __syncthreads();

  // ---- M = sum_o w_o*(2^s * L_o + log(tr_o) I); X = sym(M) * 2^-6 ----
  const float lscale = (float)(1 << SQRT_LEVELS);
  float* Msh = sT;            // slot 0 (NS T-buffers are dead now)
  float* Xe  = sT + MSZ;      // slot 1
  for (int j = tid; j < 1024; j += 256) {
    const int i = j >> 5, jj = j & 31;
    float acc = 0.f;
    #pragma unroll
    for (int m = 0; m < NOPS; ++m)
      acc += wv[m] * (lscale * Zalt[m*MSZ + i*LDM + jj] + ((i == jj) ? ltr6[m] : 0.f));
    Msh[i*LDM + jj] = acc;
  }
  __syncthreads();
  const float escale = 1.f / (float)(1 << EXP_SQUARINGS);
  for (int j = tid; j < 1024; j += 256) {
    const int i = j >> 5, jj = j & 31;
    Xe[i*LDM + jj] = 0.5f*(Msh[i*LDM + jj] + Msh[jj*LDM + i]) * escale;
  }
  __syncthreads();

  // ---- expm: degree-8 Horner Taylor + 6 squarings ----
  float* Hc = Ycur;  float* Ha = Yalt;     // reuse slot-0 regions
  for (int j = tid; j < 1024; j += 256) {  // H8 = I + X/8
    const int i = j >> 5, jj = j & 31;
    Hc[i*LDM + jj] = 0.125f*Xe[i*LDM + jj] + ((i == jj) ? 1.f : 0.f);
  }
  __syncthreads();
  for (int k = 7; k >= 1; --k) {           // H = I + (1/k) X@H
    if (wave < 4) {
      const int trow = (wave >> 1)*16, tcol = (wave & 1)*16;
      v8f acc = mm_acc<32>(Xe, LDM, Hc, LDM, trow, tcol, lane, false);
      store_affine(Ha, LDM, trow, tcol, lane, acc, 1.f/(float)k, 1.f);
    }
    __syncthreads();
    float* tmp = Hc; Hc = Ha; Ha = tmp;
  }
  for (int sq = 0; sq < EXP_SQUARINGS; ++sq) {  // F = F@F
    if (wave < 4) {
      const int trow = (wave >> 1)*16, tcol = (wave & 1)*16;
      v8f acc = mm_acc<32>(Hc, LDM, Hc, LDM, trow, tcol, lane, false);
      store_plain(Ha, LDM, trow, tcol, lane, acc);
    }
    __syncthreads();
    float* tmp = Hc; Hc = Ha; Ha = tmp;
  }

  float* og = out + (size_t)bc * 1024;
  for (int j = tid; j < 1024; j += 256)
    og[j] = Hc[(j >> 5)*LDM + (j & 31)];
}

extern "C" void kernel_launch(void* const* d_in, const int* in_sizes, int n_in,
                              void* d_out, int out_size, void* d_ws, size_t ws_size,
                              hipStream_t stream) {
  (void)in_sizes; (void)n_in; (void)out_size; (void)d_ws; (void)ws_size;
  const float* x  = (const float*)d_in[0];
  const float* W  = (const float*)d_in[1];
  const float* wv = (const float*)d_in[2];
  float* out = (float*)d_out;
  hipLaunchKernelGGL(mixedop_spd_kernel, dim3(Bn*Cn), dim3(256), 0, stream,
                     x, W, wv, out);
}